// PCGNN_26087631356719
// MI455X (gfx1250) — compile-verified
//
#include <hip/hip_runtime.h>
#include <hip/hip_bf16.h>

typedef __attribute__((ext_vector_type(2))) float v2f;
typedef __attribute__((ext_vector_type(8))) float v8f;

#define N_SRC 60000
#define N_DST 30000
#define DEG   32
#define FDIM  128
#define EDIM  64
#define KSEL  16

// ---------------------------------------------------------------------------
// Kernel 1: one wave32 per dst node.
//   score_logits = x[node] @ W_dist + b_dist
//   nb_score     = x[neighbors] @ W_dist[:,0] + b_dist[0]
//   select 16 neighbors with smallest |sigmoid(s0)-sigmoid(nb_score)|
//   h_neigh      = mean of selected rows  -> workspace
// ---------------------------------------------------------------------------
__global__ __launch_bounds__(256) void pcgnn_gather(
    const float* __restrict__ x, const int* __restrict__ neighbors,
    const float* __restrict__ W_dist, const float* __restrict__ b_dist,
    float* __restrict__ score_out,   // [N_DST,2]
    float* __restrict__ h_neigh)     // [N_DST,128]
{
    const int wid  = threadIdx.x >> 5;
    const int lane = threadIdx.x & 31;
    const int node = blockIdx.x * 8 + wid;
    if (node >= N_DST) return;                      // wave-uniform

    // ---- score_logits (cooperative 128-dot, both columns) ----
    const float4 xv = ((const float4*)(x + (size_t)node * FDIM))[lane];
    const int f0 = lane * 4;
    float p0 = xv.x * W_dist[(f0+0)*2+0] + xv.y * W_dist[(f0+1)*2+0]
             + xv.z * W_dist[(f0+2)*2+0] + xv.w * W_dist[(f0+3)*2+0];
    float p1 = xv.x * W_dist[(f0+0)*2+1] + xv.y * W_dist[(f0+1)*2+1]
             + xv.z * W_dist[(f0+2)*2+1] + xv.w * W_dist[(f0+3)*2+1];
    #pragma unroll
    for (int m = 16; m >= 1; m >>= 1) {
        p0 += __shfl_xor(p0, m, 32);
        p1 += __shfl_xor(p1, m, 32);
    }
    const float s0 = p0 + b_dist[0];
    const float s1 = p1 + b_dist[1];
    if (lane == 0) {
        score_out[node*2+0] = s0;
        score_out[node*2+1] = s1;
    }
    const float t = 1.0f / (1.0f + __expf(-s0));

    // ---- lane j scores neighbor j ----
    const int nj = neighbors[node * DEG + lane];
    const float* __restrict__ nrow = x + (size_t)nj * FDIM;
    __builtin_prefetch(nrow, 0, 3);
    float acc = 0.0f;
    #pragma unroll 4
    for (int it = 0; it < 32; ++it) {
        const float4 u = ((const float4*)nrow)[it];
        const int g = it * 4;
        acc += u.x * W_dist[(g+0)*2] + u.y * W_dist[(g+1)*2]
             + u.z * W_dist[(g+2)*2] + u.w * W_dist[(g+3)*2];
    }
    const float sg   = 1.0f / (1.0f + __expf(-(acc + b_dist[0])));
    const float diff = fabsf(t - sg);

    // ---- rank = #neighbors strictly better (index tie-break) -> top-16 ----
    int rank = 0;
    for (int i = 0; i < 32; ++i) {
        const float di = __shfl(diff, i, 32);
        rank += (di < diff) || (di == diff && i < lane);
    }
    const unsigned long long m = __ballot(rank < KSEL);

    // ---- cooperative mean of the 16 selected rows ----
    float a0 = 0.f, a1 = 0.f, a2 = 0.f, a3 = 0.f;
    for (int j = 0; j < 32; ++j) {
        if ((m >> j) & 1ull) {
            const int r = __shfl(nj, j, 32);
            const float* __restrict__ row = x + (size_t)r * FDIM;
            a0 += row[lane];
            a1 += row[lane + 32];
            a2 += row[lane + 64];
            a3 += row[lane + 96];
        }
    }
    const float inv = 1.0f / (float)KSEL;
    float* __restrict__ hn = h_neigh + (size_t)node * FDIM;
    hn[lane     ] = a0 * inv;
    hn[lane + 32] = a1 * inv;
    hn[lane + 64] = a2 * inv;
    hn[lane + 96] = a3 * inv;
}

// ---------------------------------------------------------------------------
// Kernel 2: fused SAGE layer via V_WMMA_F32_16X16X4_F32.
//   h      = relu(Xdst @ W_self + Hn @ W_neigh + b_sage)   (both GEMMs share C)
//   logits = h @ W_out + b_out                             (shfl reduction)
// One wave per 16-row tile.  Weights staged in LDS *pair-interleaved*:
//   sW[(k2*64 + col)*2 + j] = W[(2*k2 + j)*64 + col]
// so each lane's B operand {W[kb][col], W[kb+1][col]} is one aligned
// ds_load_b64 straight into the WMMA VGPR pair (no repack movs).
// ---------------------------------------------------------------------------
__global__ __launch_bounds__(256) void pcgnn_sage_wmma(
    const float* __restrict__ x,
    const float* __restrict__ h_neigh,
    const float* __restrict__ W_self,
    const float* __restrict__ W_neigh,
    const float* __restrict__ b_sage,
    const float* __restrict__ W_out,
    const float* __restrict__ b_out,
    float* __restrict__ logits)      // [N_DST,2]
{
    __shared__ float sWs[FDIM * EDIM];   // 32 KB, pair-interleaved
    __shared__ float sWn[FDIM * EDIM];   // 32 KB, pair-interleaved
    for (int i = threadIdx.x; i < FDIM * EDIM; i += 256) {
        const int kRow = i >> 6;         // 0..127
        const int col  = i & 63;
        const int dst  = (((kRow >> 1) << 6) + col) * 2 + (kRow & 1);
        sWs[dst] = W_self[i];
        sWn[dst] = W_neigh[i];
    }
    __syncthreads();

    const int wid  = threadIdx.x >> 5;
    const int lane = threadIdx.x & 31;
    const int tile = blockIdx.x * 8 + wid;
    if (tile >= N_DST / 16) return;                 // wave-uniform, EXEC stays full
    const int rowBase = tile * 16;

    const int mrow = rowBase + (lane & 15);
    const int kk   = (lane >> 4) * 2;               // lanes 0-15: K 0,1 ; lanes 16-31: K 2,3
    const int col  = lane & 15;

    const float* __restrict__ xrow = x       + (size_t)mrow * FDIM;
    const float* __restrict__ hrow = h_neigh + (size_t)mrow * FDIM;

    const v8f zero = {0.f, 0.f, 0.f, 0.f, 0.f, 0.f, 0.f, 0.f};
    v8f c[4];
    #pragma unroll
    for (int nt = 0; nt < 4; ++nt) c[nt] = zero;

    #pragma unroll 4
    for (int k = 0; k < FDIM / 4; ++k) {
        const int kb = k * 4 + kk;                  // even -> 8B-aligned float2
        const int k2 = kb >> 1;                     // pair row in LDS image
        const v2f ax = *(const v2f*)(xrow + kb);
        const v2f ah = *(const v2f*)(hrow + kb);
        #pragma unroll
        for (int nt = 0; nt < 4; ++nt) {
            const int cc = nt * 16 + col;
            const v2f bs = *(const v2f*)(sWs + ((k2 << 6) + cc) * 2);
            const v2f bn = *(const v2f*)(sWn + ((k2 << 6) + cc) * 2);
            c[nt] = __builtin_amdgcn_wmma_f32_16x16x4_f32(
                        false, ax, false, bs, (short)0, c[nt], false, false);
            c[nt] = __builtin_amdgcn_wmma_f32_16x16x4_f32(
                        false, ah, false, bn, (short)0, c[nt], false, false);
        }
    }

    // ---- epilogue: bias + relu, 64->2 projection, 16-lane reduction ----
    float bsg[4], w0[4], w1[4];
    #pragma unroll
    for (int nt = 0; nt < 4; ++nt) {
        const int cc = nt * 16 + col;
        bsg[nt] = b_sage[cc];
        w0[nt]  = W_out[cc * 2 + 0];
        w1[nt]  = W_out[cc * 2 + 1];
    }
    const float bo0 = b_out[0], bo1 = b_out[1];

    #pragma unroll
    for (int r = 0; r < 8; ++r) {
        float s0 = 0.f, s1 = 0.f;
        #pragma unroll
        for (int nt = 0; nt < 4; ++nt) {
            float h = c[nt][r] + bsg[nt];
            h = fmaxf(h, 0.f);
            s0 += h * w0[nt];
            s1 += h * w1[nt];
        }
        #pragma unroll
        for (int m = 8; m >= 1; m >>= 1) {       // reduce within each 16-lane half
            s0 += __shfl_xor(s0, m, 32);
            s1 += __shfl_xor(s1, m, 32);
        }
        if ((lane & 15) == 0) {                  // lane 0 -> M=r, lane 16 -> M=r+8
            const int row = rowBase + r + ((lane >> 4) << 3);
            logits[row * 2 + 0] = s0 + bo0;
            logits[row * 2 + 1] = s1 + bo1;
        }
    }
}

extern "C" void kernel_launch(void* const* d_in, const int* in_sizes, int n_in,
                              void* d_out, int out_size, void* d_ws, size_t ws_size,
                              hipStream_t stream) {
    const float* x       = (const float*)d_in[0];
    const int*   nb      = (const int*)  d_in[1];
    const float* W_dist  = (const float*)d_in[2];
    const float* b_dist  = (const float*)d_in[3];
    const float* W_self  = (const float*)d_in[4];
    const float* W_neigh = (const float*)d_in[5];
    const float* b_sage  = (const float*)d_in[6];
    const float* W_out   = (const float*)d_in[7];
    const float* b_out   = (const float*)d_in[8];

    float* out     = (float*)d_out;
    float* logits  = out;                 // [30000,2] first in tuple order
    float* score   = out + N_DST * 2;     // [30000,2] second
    float* h_neigh = (float*)d_ws;        // [30000,128] = 15.4 MB scratch

    pcgnn_gather<<<N_DST / 8, 256, 0, stream>>>(x, nb, W_dist, b_dist,
                                                score, h_neigh);

    const int tiles  = N_DST / 16;            // 1875
    const int blocks = (tiles + 7) / 8;       // 235
    pcgnn_sage_wmma<<<blocks, 256, 0, stream>>>(x, h_neigh, W_self, W_neigh,
                                                b_sage, W_out, b_out, logits);
}